// RGCN_47064251629674
// MI455X (gfx1250) — compile-verified
//
#include <hip/hip_runtime.h>
#include <hip/hip_bf16.h>

// ---------------------------------------------------------------------------
// RGCN (2-layer, 3-graph GCN stack) for MI455X / gfx1250, wave32.
//
// Structure:
//   deg/dinv  : f32 atomics + rsqrt  (self loop folded: deg = indeg + 1)
//   GEMM      : V_WMMA_F32_16X16X4_F32, one wave per 16x16 output tile
//   scatter   : one wave per (edge | self-loop) item; lane holds F/32 floats;
//               hardware global_atomic_add_f32 into L2-resident accumulator
// ---------------------------------------------------------------------------

typedef __attribute__((ext_vector_type(2))) float v2f;
typedef __attribute__((ext_vector_type(8))) float v8f;

#define WAVE 32

// ---------------------------------------------------------------------------
// 1) degree accumulation: deg[g*N + dst] += 1 for every edge of every graph
// ---------------------------------------------------------------------------
__global__ void deg_kernel(const int* __restrict__ edges, float* __restrict__ deg,
                           int E, int Nn) {
    int idx = blockIdx.x * blockDim.x + threadIdx.x;
    if (idx >= 3 * E) return;
    int g = idx / E;
    int e = idx - g * E;
    // edges layout [L,2,E]: dst of graph g at offset g*2E + E + e
    int d = edges[(size_t)g * 2 * E + E + e];
    unsafeAtomicAdd(&deg[(size_t)g * Nn + d], 1.0f);
}

// dinv = rsqrt(deg + 1)   (the +1 is the self loop; deg+1 >= 1 always)
__global__ void dinv_kernel(float* __restrict__ deg, int total) {
    int idx = blockIdx.x * blockDim.x + threadIdx.x;
    if (idx >= total) return;
    deg[idx] = rsqrtf(deg[idx] + 1.0f);
}

// ---------------------------------------------------------------------------
// 2) fp32 WMMA GEMM:  C[M,NOUT] = A[M,K] @ B[K,NOUT]
//    One wave computes one 16x16 tile; K-loop in steps of 4 via
//    V_WMMA_F32_16X16X4_F32.
//    A 16x4 layout : lanes 0-15 rows M=0..15 {K=k0,k0+1}; lanes 16-31 {k0+2,k0+3}
//    B 4x16 layout : lanes 0-15 cols N=0..15 rows {k0,k0+1}; lanes 16-31 {k0+2,k0+3}
//    C/D layout    : VGPR r -> lanes 0-15: (M=r, N=lane); lanes 16-31: (M=8+r)
//    M must be a multiple of 16 (50000 = 3125*16).
// ---------------------------------------------------------------------------
template <int K, int NOUT>
__global__ void gemm_wmma_f32(const float* __restrict__ A,
                              const float* __restrict__ B,
                              float* __restrict__ C, int M) {
    int wave = (blockIdx.x * blockDim.x + threadIdx.x) / WAVE;
    int lane = threadIdx.x & (WAVE - 1);
    const int tilesN = NOUT / 16;
    int tm = wave / tilesN;
    int tn = wave - tm * tilesN;
    if (tm * 16 >= M) return;  // wave-uniform; EXEC stays all-ones for WMMA

    int half = lane >> 4;   // 0 for lanes 0-15, 1 for lanes 16-31
    int l16  = lane & 15;

    const float* Abase = A + (size_t)(tm * 16 + l16) * K + 2 * half;
    const float* Bbase = B + (size_t)(2 * half) * NOUT + tn * 16 + l16;

    v8f acc = {};
#pragma unroll 4
    for (int k0 = 0; k0 < K; k0 += 4) {
        v2f a, b;
        a.x = Abase[k0];
        a.y = Abase[k0 + 1];
        b.x = Bbase[(size_t)k0 * NOUT];
        b.y = Bbase[(size_t)(k0 + 1) * NOUT];
        acc = __builtin_amdgcn_wmma_f32_16x16x4_f32(
            /*neg_a=*/false, a, /*neg_b=*/false, b,
            /*c_mod=*/(short)0, acc, /*reuse_a=*/false, /*reuse_b=*/false);
    }

    float* Crow = C + (size_t)(tm * 16 + half * 8) * NOUT + tn * 16 + l16;
#pragma unroll
    for (int r = 0; r < 8; r++) Crow[(size_t)r * NOUT] = acc[r];
}

// ---------------------------------------------------------------------------
// 3) scatter-add: one wave per item. Items 0..E-1 are edges (s=src,d=dst);
//    items E..E+N-1 are self loops (s=d=item-E) -- mirrors the reference's
//    concatenation. Lane holds VEC=F/32 consecutive floats of the row.
// ---------------------------------------------------------------------------
template <int F, int VEC>
__global__ void scatter_kernel(const float* __restrict__ xw,
                               const float* __restrict__ dinv,
                               const int* __restrict__ src,
                               const int* __restrict__ dst,
                               int E, int Nn, float* __restrict__ out) {
    int wave = (blockIdx.x * blockDim.x + threadIdx.x) / WAVE;
    int lane = threadIdx.x & (WAVE - 1);
    if (wave >= E + Nn) return;  // wave-uniform

    int s, d;
    if (wave < E) {
        s = src[wave];
        d = dst[wave];
    } else {
        s = d = wave - E;
    }
    float norm = dinv[s] * dinv[d];

    const float* xr = xw + (size_t)s * F + lane * VEC;
    float* orow     = out + (size_t)d * F + lane * VEC;
#pragma unroll
    for (int v = 0; v < VEC; v++) {
        unsafeAtomicAdd(&orow[v], xr[v] * norm);
    }
}

// ---------------------------------------------------------------------------
// 4) fused bias sum (over the 3 convs) + optional ReLU
// ---------------------------------------------------------------------------
template <int F, bool RELU>
__global__ void bias_act_kernel(float* __restrict__ h, const float* __restrict__ b,
                                int total) {
    int idx = blockIdx.x * blockDim.x + threadIdx.x;
    if (idx >= total) return;
    int f = idx & (F - 1);
    float bias = b[f] + b[F + f] + b[2 * F + f];
    float v = h[idx] + bias;
    if (RELU) v = v > 0.0f ? v : 0.0f;
    h[idx] = v;
}

// ---------------------------------------------------------------------------
// launch
// ---------------------------------------------------------------------------
extern "C" void kernel_launch(void* const* d_in, const int* in_sizes, int n_in,
                              void* d_out, int out_size, void* d_ws, size_t ws_size,
                              hipStream_t stream) {
    const float* x  = (const float*)d_in[0];   // [N, 256]
    const int* edges = (const int*)d_in[1];    // [3, 2, E]
    const float* W1 = (const float*)d_in[2];   // [3, 256, 128]
    const float* b1 = (const float*)d_in[3];   // [3, 128]
    const float* W2 = (const float*)d_in[4];   // [3, 128, 64]
    const float* b2 = (const float*)d_in[5];   // [3, 64]
    float* out = (float*)d_out;                // [N, 64]

    const int LG = 3, Fin = 256, Fhid = 128, Fout = 64;
    const int Nn = in_sizes[0] / Fin;          // 50000
    const int E  = in_sizes[1] / (LG * 2);     // 1600000

    // workspace carve-up (all fp32):
    //   dinv : LG*Nn              (~0.6 MB)
    //   xw   : Nn*Fhid            (~25.6 MB, reused as hw in layer 2)
    //   h    : Nn*Fhid            (~25.6 MB)
    float* dinv = (float*)d_ws;
    float* xw   = dinv + (size_t)LG * Nn;
    float* h    = xw + (size_t)Nn * Fhid;

    hipMemsetAsync(dinv, 0, sizeof(float) * (size_t)LG * Nn, stream);
    hipMemsetAsync(h, 0, sizeof(float) * (size_t)Nn * Fhid, stream);
    hipMemsetAsync(out, 0, sizeof(float) * (size_t)Nn * Fout, stream);

    // degrees -> dinv
    {
        int total = LG * E;
        deg_kernel<<<(total + 255) / 256, 256, 0, stream>>>(edges, dinv, E, Nn);
        int t2 = LG * Nn;
        dinv_kernel<<<(t2 + 255) / 256, 256, 0, stream>>>(dinv, t2);
    }

    // ---- layer 1: h = relu( sum_i scatter(x @ W1[i]) + sum_i b1[i] ) ----
    for (int i = 0; i < LG; i++) {
        const int* srcp = edges + (size_t)i * 2 * E;
        const int* dstp = srcp + E;

        int waves = (Nn / 16) * (Fhid / 16);
        long thr = (long)waves * WAVE;
        gemm_wmma_f32<256, 128><<<(unsigned)((thr + 255) / 256), 256, 0, stream>>>(
            x, W1 + (size_t)i * Fin * Fhid, xw, Nn);

        long items = (long)E + Nn;
        long sthr = items * WAVE;
        scatter_kernel<128, 4><<<(unsigned)((sthr + 255) / 256), 256, 0, stream>>>(
            xw, dinv + (size_t)i * Nn, srcp, dstp, E, Nn, h);
    }
    {
        int total = Nn * Fhid;
        bias_act_kernel<128, true><<<(total + 255) / 256, 256, 0, stream>>>(h, b1, total);
    }

    // ---- layer 2: out = sum_i scatter(h @ W2[i]) + sum_i b2[i] ----
    for (int i = 0; i < LG; i++) {
        const int* srcp = edges + (size_t)i * 2 * E;
        const int* dstp = srcp + E;

        int waves = (Nn / 16) * (Fout / 16);
        long thr = (long)waves * WAVE;
        gemm_wmma_f32<128, 64><<<(unsigned)((thr + 255) / 256), 256, 0, stream>>>(
            h, W2 + (size_t)i * Fhid * Fout, xw, Nn);

        long items = (long)E + Nn;
        long sthr = items * WAVE;
        scatter_kernel<64, 2><<<(unsigned)((sthr + 255) / 256), 256, 0, stream>>>(
            xw, dinv + (size_t)i * Nn, srcp, dstp, E, Nn, out);
    }
    {
        int total = Nn * Fout;
        bias_act_kernel<64, false><<<(total + 255) / 256, 256, 0, stream>>>(out, b2, total);
    }
}